// PersistenceLandscapeLayer_39556648796335
// MI455X (gfx1250) — compile-verified
//
#include <hip/hip_runtime.h>
#include <stdint.h>

#define P_PAIRS 4096
#define T_SEQ   50
#define K_MAX   5
#define NDIM    2
#define THREADS 320   // 10 waves (wave32); wave w owns t in [5w, 5w+5) -> exactly 50 t's

// Branchless sorted-descending top-5 insert: 5 max/min compare-swap steps.
// Lists are invariantly >= 0 (init 0, updates are max()), so inserting a
// negative value is a no-op -- callers may skip clamping tent values to 0.
__device__ __forceinline__ void insert5(float (&L)[5], float v) {
#pragma unroll
  for (int k = 0; k < 5; ++k) {
    float hi = fmaxf(L[k], v);
    v        = fminf(L[k], v);
    L[k]     = hi;
  }
}

// wave32 merge of per-lane sorted top-5 lists -> lane 0 writes sorted top-5.
__device__ __forceinline__ void merge_write5(float (&L)[5], float* dst, int lane) {
  float h0 = L[0], h1 = L[1], h2 = L[2], h3 = L[3], h4 = L[4];
#pragma unroll
  for (int r = 0; r < K_MAX; ++r) {
    float m = h0;
#pragma unroll
    for (int off = 16; off; off >>= 1) m = fmaxf(m, __shfl_xor(m, off, 32));
    unsigned long long ball = __ballot(h0 == m);
    int src = (int)__ffsll(ball) - 1;          // lowest lane holding the max pops it
    if (lane == src) { h0 = h1; h1 = h2; h2 = h3; h3 = h4; h4 = 0.0f; }
    if (lane == 0) dst[r] = m;
  }
}

__global__ __launch_bounds__(THREADS)
void landscape_topk_kernel(const float* __restrict__ births,
                           const float* __restrict__ deaths,
                           const int*   __restrict__ dims,
                           float* __restrict__ out)
{
  __shared__ float  ldsB[P_PAIRS];
  __shared__ float  ldsD[P_PAIRS];
  __shared__ int    ldsM[P_PAIRS];
  __shared__ float2 pk[P_PAIRS];   // dim0 packed from front, dim1 packed from back
  __shared__ int    cnt0, cnt1;

  const int b    = blockIdx.x;
  const int tid  = threadIdx.x;
  const int lane = tid & 31;
  const int wave = tid >> 5;

  if (tid == 0) { cnt0 = 0; cnt1 = 0; }

  // ---- Stage this batch row into LDS via CDNA5 async DMA (ASYNCcnt path) ----
  {
    const size_t rowOff = (size_t)b * P_PAIRS;
    const float4* gB = (const float4*)(births + rowOff);
    const float4* gD = (const float4*)(deaths + rowOff);
    const int4*   gM = (const int4*)(dims   + rowOff);
    for (int idx = tid; idx < P_PAIRS / 4; idx += THREADS) {
      // Low 32 bits of a flat __shared__ pointer == workgroup-relative LDS byte offset
      uint32_t laB = (uint32_t)(uintptr_t)&ldsB[idx * 4];
      uint32_t laD = (uint32_t)(uintptr_t)&ldsD[idx * 4];
      uint32_t laM = (uint32_t)(uintptr_t)&ldsM[idx * 4];
      uint64_t gaB = (uint64_t)(uintptr_t)(gB + idx);
      uint64_t gaD = (uint64_t)(uintptr_t)(gD + idx);
      uint64_t gaM = (uint64_t)(uintptr_t)(gM + idx);
      asm volatile("global_load_async_to_lds_b128 %0, %1, off" :: "v"(laB), "v"(gaB) : "memory");
      asm volatile("global_load_async_to_lds_b128 %0, %1, off" :: "v"(laD), "v"(gaD) : "memory");
      asm volatile("global_load_async_to_lds_b128 %0, %1, off" :: "v"(laM), "v"(gaM) : "memory");
    }
    asm volatile("s_wait_asynccnt 0" ::: "memory");
    __syncthreads();
  }

  // ---- Compact pairs by dim (warp-aggregated LDS atomics; order-free for top-k) ----
  for (int p = tid; p < P_PAIRS; p += THREADS) {
    const int   dm = ldsM[p];
    const float2 e = make_float2(ldsB[p], ldsD[p]);
    const unsigned m0 = (unsigned)__ballot(dm == 0);
    const unsigned m1 = (unsigned)__ballot(dm != 0);
    int slot;
    if (dm == 0) {
      int leader = __ffs(m0) - 1, base = 0;
      if (lane == leader) base = atomicAdd(&cnt0, __popc(m0));
      base = __shfl(base, leader, 32);
      slot = base + __popc(m0 & ((1u << lane) - 1u));
    } else {
      int leader = __ffs(m1) - 1, base = 0;
      if (lane == leader) base = atomicAdd(&cnt1, __popc(m1));
      base = __shfl(base, leader, 32);
      slot = (P_PAIRS - 1) - (base + __popc(m1 & ((1u << lane) - 1u)));
    }
    pk[slot] = e;
  }
  __syncthreads();
  const int n0 = cnt0;
  const int n1 = cnt1;

  const int tBase = wave * 5;
  float tv[5];
#pragma unroll
  for (int j = 0; j < 5; ++j) tv[j] = (float)(tBase + j) * 0.02f;

  float L0[5][5], L1[5][5];
#pragma unroll
  for (int j = 0; j < 5; ++j)
#pragma unroll
    for (int k = 0; k < 5; ++k) { L0[j][k] = 0.0f; L1[j][k] = 0.0f; }

  // ---- Scan: one ds_load_b64 + ONE insert chain per (pair, t).
  //      No clamp needed: negative tent values cannot enter a >=0 list. ----
  for (int p = lane; p < n0; p += 32) {
    const float2 e = pk[p];
#pragma unroll
    for (int j = 0; j < 5; ++j) {
      insert5(L0[j], fminf(tv[j] - e.x, e.y - tv[j]));
    }
  }
  for (int p = lane; p < n1; p += 32) {
    const float2 e = pk[(P_PAIRS - 1) - p];
#pragma unroll
    for (int j = 0; j < 5; ++j) {
      insert5(L1[j], fminf(tv[j] - e.x, e.y - tv[j]));
    }
  }

  // ---- Cross-lane merge + write: out[((b*2 + d)*50 + t)*5 + r] ----
#pragma unroll
  for (int j = 0; j < 5; ++j) {
    const int ti = tBase + j;
    merge_write5(L0[j], out + (((size_t)b * NDIM + 0) * T_SEQ + ti) * K_MAX, lane);
    merge_write5(L1[j], out + (((size_t)b * NDIM + 1) * T_SEQ + ti) * K_MAX, lane);
  }
}

extern "C" void kernel_launch(void* const* d_in, const int* in_sizes, int n_in,
                              void* d_out, int out_size, void* d_ws, size_t ws_size,
                              hipStream_t stream) {
  const float* births = (const float*)d_in[0];
  const float* deaths = (const float*)d_in[1];
  const int*   dims   = (const int*)d_in[2];
  float* out = (float*)d_out;
  (void)n_in; (void)out_size; (void)d_ws; (void)ws_size;
  const int B = in_sizes[0] / P_PAIRS;   // 256 in the reference setup
  landscape_topk_kernel<<<B, THREADS, 0, stream>>>(births, deaths, dims, out);
}